// BoundaryGCN_87986700026232
// MI455X (gfx1250) — compile-verified
//
#include <hip/hip_runtime.h>

typedef __attribute__((ext_vector_type(2))) float v2f;
typedef __attribute__((ext_vector_type(4))) float v4f;
typedef __attribute__((ext_vector_type(8))) float v8f;

#define EMBD 128
#define LN_EPS 1e-5f

// ---------------------------------------------------------------------------
// Per-node init: inv[i] = rsqrt(degree[i]+1), cnt[i] = (int)degree[i]
__global__ __launch_bounds__(256) void node_init_kernel(const float* __restrict__ deg,
                                                        float* __restrict__ inv,
                                                        int* __restrict__ cnt, int n) {
  int i = blockIdx.x * 256 + threadIdx.x;
  if (i < n) {
    float d = deg[i];
    inv[i] = rsqrtf(d + 1.0f);
    cnt[i] = (int)(d + 0.5f);
  }
}

__global__ __launch_bounds__(256) void zero_int_kernel(int* __restrict__ p, int n) {
  int i = blockIdx.x * 256 + threadIdx.x;
  if (i < n) p[i] = 0;
}

// ---------------------------------------------------------------------------
// Single-block exclusive scan of cnt[0..N) -> row_ptr[0..N], row_ptr[N]=total.
__global__ __launch_bounds__(1024) void scan_kernel(const int* __restrict__ cnt,
                                                    int* __restrict__ row_ptr, int N) {
  __shared__ int part[1024];
  const int tid = threadIdx.x;
  const int chunk = (N + 1023) / 1024;
  int beg = tid * chunk;
  int end = beg + chunk;
  if (end > N) end = N;
  int s = 0;
  for (int i = beg; i < end; ++i) s += cnt[i];
  part[tid] = s;
  __syncthreads();
  // inclusive Hillis-Steele scan over the 1024 partials
  for (int off = 1; off < 1024; off <<= 1) {
    int v = (tid >= off) ? part[tid - off] : 0;
    __syncthreads();
    part[tid] += v;
    __syncthreads();
  }
  int run = (tid == 0) ? 0 : part[tid - 1];
  for (int i = beg; i < end; ++i) {
    row_ptr[i] = run;
    run += cnt[i];
  }
  if (tid == 1023) row_ptr[N] = part[1023];
}

// Fill CSR: for each edge e, csr[row_ptr[dst] + rank] = src. (ranks via int atomics)
__global__ __launch_bounds__(256) void fill_kernel(const int* __restrict__ ei,
                                                   const int* __restrict__ row_ptr,
                                                   int* __restrict__ cnt,
                                                   int* __restrict__ csr, int E) {
  int e = blockIdx.x * 256 + threadIdx.x;
  if (e < E) {
    int d = ei[E + e];
    int pos = row_ptr[d] + atomicAdd(&cnt[d], 1);
    csr[pos] = ei[e];
  }
}

// Insertion-sort each dst segment (avg len ~16) -> deterministic gather order.
__global__ __launch_bounds__(256) void sort_kernel(const int* __restrict__ row_ptr,
                                                   int* __restrict__ csr, int N) {
  int d = blockIdx.x * 256 + threadIdx.x;
  if (d >= N) return;
  int beg = row_ptr[d], end = row_ptr[d + 1];
  for (int i = beg + 1; i < end; ++i) {
    int key = csr[i];
    int j = i - 1;
    while (j >= beg && csr[j] > key) {
      csr[j + 1] = csr[j];
      --j;
    }
    csr[j + 1] = key;
  }
}

// ---------------------------------------------------------------------------
// Repack W[128 x NCOLS] so each lane's B fragment (2 floats from rows k,k+1)
// is one contiguous 8-byte load:
// Wp[ ((k>>2)*2 + half)*NCOLS + col ] (as float2) = { W[k0+2h][col], W[k0+2h+1][col] }
__global__ __launch_bounds__(256) void pack_w_kernel(const float* __restrict__ W,
                                                     float* __restrict__ Wp, int NCOLS) {
  int idx = blockIdx.x * 256 + threadIdx.x;
  if (idx >= EMBD * NCOLS) return;
  int k = idx / NCOLS, col = idx % NCOLS;
  int g = k >> 2, kk = k & 3, half = kk >> 1, lo = kk & 1;
  Wp[(((g * 2) + half) * NCOLS + col) * 2 + lo] = W[idx];
}

// ---------------------------------------------------------------------------
// C[nrows x NCOLS] = act(A[nrows x 128] @ W[128 x NCOLS] + bias (+ res))
// Block = 256 threads = 8 waves; wave w computes rows [blk*128 + w*16, +16).
// fp32 WMMA (V_WMMA_F32_16X16X4_F32), K swept 4 at a time. Wp is the packed W.
template <int NCOLS, bool RELU, bool RES>
__global__ __launch_bounds__(256) void gemm_kernel(const float* __restrict__ A,
                                                   const float* __restrict__ Wp,
                                                   const float* __restrict__ bias,
                                                   const float* __restrict__ res,
                                                   float* __restrict__ C, int nrows) {
  const int tid  = threadIdx.x;
  const int wave = tid >> 5;
  const int lane = tid & 31;
  const int half = lane >> 4;   // 0: lanes 0-15, 1: lanes 16-31
  const int lrow = lane & 15;
  const int m0   = blockIdx.x * 128 + wave * 16;

  v8f acc[NCOLS / 16] = {};

  int arow = m0 + lrow;
  if (arow >= nrows) arow = nrows - 1;
  const float* __restrict__ Arow = A + (long)arow * EMBD;

  for (int k0 = 0; k0 < EMBD; k0 += 4) {
    // A 16x4 f32 layout: vgpr0 = K(k0 + 2*half), vgpr1 = K(k0 + 2*half + 1)
    v2f a = *(const v2f*)(Arow + k0 + 2 * half);
    const float* __restrict__ Wk = Wp + (((k0 >> 2) * 2 + half) * NCOLS) * 2;
#pragma unroll
    for (int nt = 0; nt < NCOLS / 16; ++nt) {
      const int col = nt * 16 + lrow;
      v2f b = *(const v2f*)(Wk + col * 2);
      acc[nt] = __builtin_amdgcn_wmma_f32_16x16x4_f32(
          false, a, false, b, (short)0, acc[nt], false, false);
    }
  }

  // C layout: vgpr j -> row (m0 + 8*half + j), col = nt*16 + lrow
#pragma unroll
  for (int nt = 0; nt < NCOLS / 16; ++nt) {
    const int col = nt * 16 + lrow;
    const float bb = bias[col];
#pragma unroll
    for (int j = 0; j < 8; ++j) {
      const int row = m0 + half * 8 + j;
      if (row < nrows) {
        float v = acc[nt][j] + bb;
        if (RES) v += res[(long)row * NCOLS + col];
        if (RELU) v = fmaxf(v, 0.0f);
        C[(long)row * NCOLS + col] = v;
      }
    }
  }
}

// ---------------------------------------------------------------------------
// Atomic-free, deterministic aggregation. One wave per dst node:
//   agg[d] = inv[d] * ( sum_{s in csr[d]} inv[s]*m[s]  +  inv[d]*m[d] )
__global__ __launch_bounds__(256) void gather_kernel(const float* __restrict__ m,
                                                     const int* __restrict__ row_ptr,
                                                     const int* __restrict__ csr,
                                                     const float* __restrict__ inv,
                                                     float* __restrict__ agg, int N) {
  const long gid  = (long)blockIdx.x * 256 + threadIdx.x;
  const long d    = gid >> 5;
  const int  lane = (int)(gid & 31);
  if (d >= N) return;
  const int beg = row_ptr[d], end = row_ptr[d + 1];
  v4f acc = {0.f, 0.f, 0.f, 0.f};
  for (int idx = beg; idx < end; ++idx) {
    int s = csr[idx];
    float c = inv[s];
    v4f v = *(const v4f*)(m + (long)s * EMBD + lane * 4);
    acc += v * c;
  }
  const float invd = inv[d];
  v4f self = *(const v4f*)(m + d * EMBD + lane * 4);
  v4f r = (acc + self * invd) * invd;
  *(v4f*)(agg + d * EMBD + lane * 4) = r;
}

// ---------------------------------------------------------------------------
// One wave per row: Y = maybe_relu(LN(X) * g + b), feature dim 128.
template <bool RELU>
__global__ __launch_bounds__(256) void ln_kernel(const float* __restrict__ X,
                                                 const float* __restrict__ g,
                                                 const float* __restrict__ b,
                                                 float* __restrict__ Y, int nrows) {
  const long gid  = (long)blockIdx.x * 256 + threadIdx.x;
  const long r    = gid >> 5;
  const int  lane = (int)(gid & 31);
  if (r >= nrows) return;
  v4f v = *(const v4f*)(X + r * EMBD + lane * 4);
  float s = v.x + v.y + v.z + v.w;
#pragma unroll
  for (int off = 16; off > 0; off >>= 1) s += __shfl_xor(s, off, 32);
  const float mean = s * (1.0f / 128.0f);
  v4f dv = v - mean;
  float q = dv.x * dv.x + dv.y * dv.y + dv.z * dv.z + dv.w * dv.w;
#pragma unroll
  for (int off = 16; off > 0; off >>= 1) q += __shfl_xor(q, off, 32);
  const float rstd = rsqrtf(q * (1.0f / 128.0f) + LN_EPS);
  v4f gg = *(const v4f*)(g + lane * 4);
  v4f bb = *(const v4f*)(b + lane * 4);
  v4f y = dv * rstd * gg + bb;
  if (RELU) {
    y.x = fmaxf(y.x, 0.f);
    y.y = fmaxf(y.y, 0.f);
    y.z = fmaxf(y.z, 0.f);
    y.w = fmaxf(y.w, 0.f);
  }
  *(v4f*)(Y + r * EMBD + lane * 4) = y;
}

// ---------------------------------------------------------------------------
extern "C" void kernel_launch(void* const* d_in, const int* in_sizes, int n_in,
                              void* d_out, int out_size, void* d_ws, size_t ws_size,
                              hipStream_t stream) {
  const float* x      = (const float*)d_in[0];
  const int*   ei     = (const int*)d_in[1];
  const float* degree = (const float*)d_in[2];
  const float* Win    = (const float*)d_in[3];
  const float* b_in   = (const float*)d_in[4];
  const float* W1     = (const float*)d_in[5];
  const float* b1     = (const float*)d_in[6];
  const float* W2     = (const float*)d_in[7];
  const float* b2     = (const float*)d_in[8];
  const float* ln_g   = (const float*)d_in[9];
  const float* ln_b   = (const float*)d_in[10];
  const float* out_g  = (const float*)d_in[11];
  const float* out_b  = (const float*)d_in[12];
  const float* Wout   = (const float*)d_in[13];
  const float* b_out  = (const float*)d_in[14];
  float* out = (float*)d_out;

  const int N = in_sizes[0] / EMBD;
  const int E = in_sizes[1] / 2;
  (void)ws_size; (void)n_in; (void)out_size;

  // ---- workspace carve -----------------------------------------------------
  char* ws = (char*)d_ws;
  size_t off = 0;
  auto carve = [&](size_t bytes) {
    void* p = ws + off;
    off = (off + bytes + 255) & ~(size_t)255;
    return p;
  };
  float* inv     = (float*)carve((size_t)N * 4);
  int*   cnt     = (int*)carve((size_t)N * 4);
  int*   row_ptr = (int*)carve((size_t)(N + 1) * 4);
  int*   csr     = (int*)carve((size_t)E * 4);
  float* WinP    = (float*)carve((size_t)EMBD * EMBD * 4);
  float* W1P     = (float*)carve((size_t)3 * EMBD * EMBD * 4);
  float* W2P     = (float*)carve((size_t)3 * EMBD * EMBD * 4);
  float* WoutP   = (float*)carve((size_t)EMBD * 64 * 4);
  float* h       = (float*)carve((size_t)N * EMBD * 4);
  float* m       = (float*)carve((size_t)N * EMBD * 4);
  float* agg     = (float*)carve((size_t)N * EMBD * 4);

  const int nblk   = (N + 255) / 256;
  const int gemm_b = (N + 127) / 128;
  const int wave_b = (N + 7) / 8;       // wave-per-node kernels
  const int pw_b   = (EMBD * EMBD + 255) / 256;

  // ---- CSR build (deterministic; reused by all 3 layers) -------------------
  node_init_kernel<<<nblk, 256, 0, stream>>>(degree, inv, cnt, N);
  scan_kernel<<<1, 1024, 0, stream>>>(cnt, row_ptr, N);
  zero_int_kernel<<<nblk, 256, 0, stream>>>(cnt, N);
  fill_kernel<<<(E + 255) / 256, 256, 0, stream>>>(ei, row_ptr, cnt, csr, E);
  sort_kernel<<<nblk, 256, 0, stream>>>(row_ptr, csr, N);

  // ---- weight repack for single-load B fragments ---------------------------
  pack_w_kernel<<<pw_b, 256, 0, stream>>>(Win, WinP, EMBD);
  for (int i = 0; i < 3; ++i) {
    pack_w_kernel<<<pw_b, 256, 0, stream>>>(W1 + (size_t)i * EMBD * EMBD,
                                            W1P + (size_t)i * EMBD * EMBD, EMBD);
    pack_w_kernel<<<pw_b, 256, 0, stream>>>(W2 + (size_t)i * EMBD * EMBD,
                                            W2P + (size_t)i * EMBD * EMBD, EMBD);
  }
  pack_w_kernel<<<(EMBD * 64 + 255) / 256, 256, 0, stream>>>(Wout, WoutP, 64);

  // ---- network -------------------------------------------------------------
  // h = relu(x @ Win + b_in)
  gemm_kernel<128, true, false><<<gemm_b, 256, 0, stream>>>(x, WinP, b_in, nullptr, h, N);

  for (int i = 0; i < 3; ++i) {
    // m = h @ W1[i] + b1[i]
    gemm_kernel<128, false, false><<<gemm_b, 256, 0, stream>>>(
        h, W1P + (size_t)i * EMBD * EMBD, b1 + i * EMBD, nullptr, m, N);
    // agg = sym-normalized neighborhood sum (atomic-free gather)
    gather_kernel<<<wave_b, 256, 0, stream>>>(m, row_ptr, csr, inv, agg, N);
    // t = agg @ W2[i] + b2[i] + h   (reuse m as t)
    gemm_kernel<128, false, true><<<gemm_b, 256, 0, stream>>>(
        agg, W2P + (size_t)i * EMBD * EMBD, b2 + i * EMBD, h, m, N);
    // h = relu(LN(t))
    ln_kernel<true><<<wave_b, 256, 0, stream>>>(m, ln_g + i * EMBD, ln_b + i * EMBD, h, N);
  }

  // final LN into m, then out = m @ Wout + b_out
  ln_kernel<false><<<wave_b, 256, 0, stream>>>(h, out_g, out_b, m, N);
  gemm_kernel<64, false, false><<<gemm_b, 256, 0, stream>>>(m, WoutP, b_out, nullptr, out, N);
}